// PSTNN_26886495273241
// MI455X (gfx1250) — compile-verified
//
#include <hip/hip_runtime.h>
#include <cstddef>
#include <cstdint>
#include <cmath>

// ---------------------------------------------------------------------------
// CDNA5 (gfx1250) structure-tensor corner-prior pipeline.
// 40-tap separable Gaussian convs run on the matrix core as banded-Toeplitz
// GEMMs (V_WMMA_F32_16X16X4_F32, full f32). Input halo tiles are staged into
// LDS via gfx1250 async global->LDS copies (ASYNCcnt / s_wait_asynccnt), and
// WMMA fragments are fed from LDS with conflict-free padding.
// ---------------------------------------------------------------------------

typedef float v2f __attribute__((ext_vector_type(2)));
typedef float v8f __attribute__((ext_vector_type(8)));

#define KTAPS   40
#define KHALF   20
#define KSTEPS  14          // K = 56 >= 16 + 40 - 1 (banded Toeplitz width)
#define WAVES   8           // waves (row tiles) per 256-thread block

struct Taps { float g[KTAPS]; };

// ---------------- async global->LDS staging (gfx1250 path) -----------------
#if defined(__has_builtin)
#if __has_builtin(__builtin_amdgcn_global_load_async_to_lds_b32) && \
    __has_builtin(__builtin_amdgcn_s_wait_asynccnt)
#define USE_ASYNC_LDS 1
#endif
#endif

typedef __attribute__((address_space(1))) int g_i32;   // prints as __device__ int*
typedef __attribute__((address_space(3))) int l_i32;   // prints as __shared__ int*

__device__ __forceinline__ void stage_b32(const float* gp, float* lp) {
#ifdef USE_ASYNC_LDS
    __builtin_amdgcn_global_load_async_to_lds_b32(
        (g_i32*)(uintptr_t)gp, (l_i32*)lp, 0, 0);
#else
    *lp = *gp;
#endif
}

__device__ __forceinline__ void stage_wait() {
#ifdef USE_ASYNC_LDS
    __builtin_amdgcn_s_wait_asynccnt(0);
#endif
    __asm__ volatile("" ::: "memory");
}

// np.pad mode='reflect' (no edge repeat); branchless so EXEC stays all-ones.
__device__ __forceinline__ int reflect_idx(int p, int n) {
    p = (p < 0) ? -p : p;
    p = (p >= n) ? (2 * n - 2 - p) : p;
    return p;
}

// Toeplitz entry T[m,k] = g[k-m] for (k-m) in [0,40), else 0; g lives in LDS.
__device__ __forceinline__ float toep_lds(const float* g, int m, int k) {
    const int t  = k - m;
    const int tc = min(max(t, 0), KTAPS - 1);
    const float v = g[tc];
    return ((unsigned)t < (unsigned)KTAPS) ? v : 0.0f;
}

// ---------------------------------------------------------------------------
// Vertical 40-tap conv, reflect boundary: out[Hout x W], Hout = Hin + 1.
// One wave -> one 16x16 tile.  C = T(16x56) * X(56x16), X staged in LDS.
// ---------------------------------------------------------------------------
__global__ void __launch_bounds__(256)
vconv_wmma(const float* __restrict__ in, float* __restrict__ out,
           int Hin, int W, int Hout, Taps taps)
{
    __shared__ float g[KTAPS];
    __shared__ float tile[WAVES][56][16];

    if (threadIdx.x < KTAPS) g[threadIdx.x] = taps.g[threadIdx.x];
    __syncthreads();

    const int w    = threadIdx.x >> 5;
    const int lane = threadIdx.x & 31;
    const int half = lane >> 4;
    const int l16  = lane & 15;
    const int n0 = blockIdx.x * 16;
    const int mt = blockIdx.y * WAVES + w;
    const int m0 = mt * 16;
    if (m0 >= Hout) return;                       // after the only barrier

    const float* src = in  + (size_t)blockIdx.z * (size_t)Hin  * W;
    float*       dst = out + (size_t)blockIdx.z * (size_t)Hout * W;

    // Stage 56x16 halo tile: coalesced (16 consecutive cols per row).
#pragma unroll
    for (int j = 0; j < 28; ++j) {                // 896 elems / 32 lanes
        const int idx = j * 32 + lane;
        const int r = idx >> 4, cc = idx & 15;
        const int grow = reflect_idx(m0 + r - KHALF, Hin);
        const int gcol = min(n0 + cc, W - 1);
        stage_b32(&src[(size_t)grow * W + gcol], &tile[w][r][cc]);
    }
    stage_wait();

    v8f c = {};
#pragma unroll
    for (int ks = 0; ks < KSTEPS; ++ks) {
        const int kb = 4 * ks + 2 * half;
        v2f a, b;
        a.x = toep_lds(g, l16, kb);
        a.y = toep_lds(g, l16, kb + 1);
        b.x = tile[w][kb][l16];
        b.y = tile[w][kb + 1][l16];
        c = __builtin_amdgcn_wmma_f32_16x16x4_f32(false, a, false, b,
                                                  (short)0, c, false, false);
    }

    const int cc = n0 + l16;
    if (cc < W) {
#pragma unroll
        for (int r = 0; r < 8; ++r) {
            const int row = m0 + r + 8 * half;    // D layout: M = r + 8*half
            if (row < Hout) dst[(size_t)row * W + cc] = c[r];
        }
    }
}

// ---------------------------------------------------------------------------
// Horizontal 40-tap conv, reflect boundary: out[H x Wout], Wout = Win + 1.
// C = X(16x56) * T^t(56x16).  X staged to LDS (row-major, 57 pad -> no bank
// conflicts on the strided A-fragment reads).
// ---------------------------------------------------------------------------
__global__ void __launch_bounds__(256)
hconv_wmma(const float* __restrict__ in, float* __restrict__ out,
           int H, int Win, int Wout, Taps taps)
{
    __shared__ float g[KTAPS];
    __shared__ float tile[WAVES][16][57];

    if (threadIdx.x < KTAPS) g[threadIdx.x] = taps.g[threadIdx.x];
    __syncthreads();

    const int w    = threadIdx.x >> 5;
    const int lane = threadIdx.x & 31;
    const int half = lane >> 4;
    const int l16  = lane & 15;
    const int n0 = blockIdx.x * 16;
    const int mt = blockIdx.y * WAVES + w;
    const int m0 = mt * 16;
    if (m0 >= H) return;

    const float* src = in  + (size_t)blockIdx.z * (size_t)H * Win;
    float*       dst = out + (size_t)blockIdx.z * (size_t)H * Wout;

    // Stage 16x56 halo tile: coalesced along rows (reflect is piecewise-contig).
#pragma unroll
    for (int j = 0; j < 28; ++j) {
        const int idx = j * 32 + lane;
        const int r = idx / 56, cc = idx % 56;
        const int grow = min(m0 + r, H - 1);
        const int gcol = reflect_idx(n0 + cc - KHALF, Win);
        stage_b32(&src[(size_t)grow * Win + gcol], &tile[w][r][cc]);
    }
    stage_wait();

    v8f c = {};
#pragma unroll
    for (int ks = 0; ks < KSTEPS; ++ks) {
        const int kb = 4 * ks + 2 * half;
        v2f a, b;
        a.x = tile[w][l16][kb];
        a.y = tile[w][l16][kb + 1];
        b.x = toep_lds(g, l16, kb);               // B[k,n] = g[k-n]
        b.y = toep_lds(g, l16, kb + 1);
        c = __builtin_amdgcn_wmma_f32_16x16x4_f32(false, a, false, b,
                                                  (short)0, c, false, false);
    }

    const int cc = n0 + l16;
    if (cc < Wout) {
#pragma unroll
        for (int r = 0; r < 8; ++r) {
            const int rr = m0 + r + 8 * half;
            if (rr < H) dst[(size_t)rr * Wout + cc] = c[r];
        }
    }
}

// ---------------------------------------------------------------------------
// Fused: horizontal blur2 of (J11,J12,J22) + structure-tensor eigen math.
// Three WMMA accumulator chains share one Toeplitz B fragment.
// ---------------------------------------------------------------------------
__global__ void __launch_bounds__(256)
hconv2_eigen_wmma(const float* __restrict__ t11, const float* __restrict__ t12,
                  const float* __restrict__ t22, float* __restrict__ sdst,
                  int H, int Win, int Wout, Taps taps)
{
    __shared__ float g[KTAPS];
    __shared__ float s11[WAVES][16][57];
    __shared__ float s12[WAVES][16][57];
    __shared__ float s22[WAVES][16][57];

    if (threadIdx.x < KTAPS) g[threadIdx.x] = taps.g[threadIdx.x];
    __syncthreads();

    const int w    = threadIdx.x >> 5;
    const int lane = threadIdx.x & 31;
    const int half = lane >> 4;
    const int l16  = lane & 15;
    const int n0 = blockIdx.x * 16;
    const int mt = blockIdx.y * WAVES + w;
    const int m0 = mt * 16;
    if (m0 >= H) return;

    const size_t ibase = (size_t)blockIdx.z * (size_t)H * Win;
    float* dst = sdst + (size_t)blockIdx.z * (size_t)H * Wout;

#pragma unroll
    for (int j = 0; j < 28; ++j) {
        const int idx = j * 32 + lane;
        const int r = idx / 56, cc = idx % 56;
        const int grow = min(m0 + r, H - 1);
        const int gcol = reflect_idx(n0 + cc - KHALF, Win);
        const size_t go = ibase + (size_t)grow * Win + gcol;
        stage_b32(&t11[go], &s11[w][r][cc]);
        stage_b32(&t12[go], &s12[w][r][cc]);
        stage_b32(&t22[go], &s22[w][r][cc]);
    }
    stage_wait();

    v8f c11 = {}, c12 = {}, c22 = {};
#pragma unroll
    for (int ks = 0; ks < KSTEPS; ++ks) {
        const int kb = 4 * ks + 2 * half;
        v2f a, b;
        b.x = toep_lds(g, l16, kb);
        b.y = toep_lds(g, l16, kb + 1);
        a.x = s11[w][l16][kb]; a.y = s11[w][l16][kb + 1];
        c11 = __builtin_amdgcn_wmma_f32_16x16x4_f32(false, a, false, b,
                                                    (short)0, c11, false, false);
        a.x = s12[w][l16][kb]; a.y = s12[w][l16][kb + 1];
        c12 = __builtin_amdgcn_wmma_f32_16x16x4_f32(false, a, false, b,
                                                    (short)0, c12, false, false);
        a.x = s22[w][l16][kb]; a.y = s22[w][l16][kb + 1];
        c22 = __builtin_amdgcn_wmma_f32_16x16x4_f32(false, a, false, b,
                                                    (short)0, c22, false, false);
    }

    const int cc = n0 + l16;
    if (cc < Wout) {
#pragma unroll
        for (int r = 0; r < 8; ++r) {
            const int rr = m0 + r + 8 * half;
            if (rr < H) {
                const float j11 = c11[r], j12 = c12[r], j22 = c22[r];
                const float d  = j11 - j22;
                const float sq = sqrtf(d * d + 4.0f * j12 * j12);
                const float l1 = 0.5f * (j11 + j22 + sq);
                const float l2 = 0.5f * (j11 + j22 - sq);
                const float corner = l1 * l2 / (l1 + l2 + 1e-5f);
                dst[(size_t)rr * Wout + cc] = corner * fmaxf(l1, l2);
            }
        }
    }
}

// ---------------------------------------------------------------------------
// Gradient + products (elementwise).
// ---------------------------------------------------------------------------
__global__ void grad_products(const float* __restrict__ img,
                              float* __restrict__ gxx, float* __restrict__ gxy,
                              float* __restrict__ gyy,
                              int H, int W, long total)
{
    long i = (long)blockIdx.x * blockDim.x + threadIdx.x;
    if (i >= total) return;
    const long nper = (long)H * W;
    const int  b = (int)(i / nper);
    const long rem = i % nper;
    const int y = (int)(rem / W);
    const int x = (int)(rem % W);
    const float* p = img + (size_t)b * nper;

    float gx = (x == 0)     ? p[(size_t)y * W + 1] - p[(size_t)y * W]
             : (x == W - 1) ? p[(size_t)y * W + W - 1] - p[(size_t)y * W + W - 2]
             : 0.5f * (p[(size_t)y * W + x + 1] - p[(size_t)y * W + x - 1]);
    float gy = (y == 0)     ? p[(size_t)W + x] - p[x]
             : (y == H - 1) ? p[(size_t)(H - 1) * W + x] - p[(size_t)(H - 2) * W + x]
             : 0.5f * (p[(size_t)(y + 1) * W + x] - p[(size_t)(y - 1) * W + x]);

    gxx[i] = gx * gx;
    gxy[i] = gx * gy;
    gyy[i] = gy * gy;
}

// ---------------------------------------------------------------------------
// Deterministic two-stage global min/max (graph-capture safe, no atomics).
// ---------------------------------------------------------------------------
__global__ void __launch_bounds__(256)
reduce_minmax_1(const float* __restrict__ s, long n,
                float* __restrict__ pmin, float* __restrict__ pmax)
{
    __shared__ float smn[256], smx[256];
    float mn = 3.402823466e+38f, mx = -3.402823466e+38f;
    for (long i = (long)blockIdx.x * blockDim.x + threadIdx.x; i < n;
         i += (long)gridDim.x * blockDim.x) {
        const float v = s[i];
        mn = fminf(mn, v);
        mx = fmaxf(mx, v);
    }
    smn[threadIdx.x] = mn; smx[threadIdx.x] = mx;
    __syncthreads();
    for (int o = 128; o > 0; o >>= 1) {
        if (threadIdx.x < o) {
            smn[threadIdx.x] = fminf(smn[threadIdx.x], smn[threadIdx.x + o]);
            smx[threadIdx.x] = fmaxf(smx[threadIdx.x], smx[threadIdx.x + o]);
        }
        __syncthreads();
    }
    if (threadIdx.x == 0) { pmin[blockIdx.x] = smn[0]; pmax[blockIdx.x] = smx[0]; }
}

__global__ void __launch_bounds__(256)
reduce_minmax_2(const float* __restrict__ pmin, const float* __restrict__ pmax,
                float* __restrict__ scal)
{
    __shared__ float smn[256], smx[256];
    const int t = threadIdx.x;
    smn[t] = fminf(pmin[t], pmin[t + 256]);
    smx[t] = fmaxf(pmax[t], pmax[t + 256]);
    __syncthreads();
    for (int o = 128; o > 0; o >>= 1) {
        if (t < o) {
            smn[t] = fminf(smn[t], smn[t + o]);
            smx[t] = fmaxf(smx[t], smx[t + o]);
        }
        __syncthreads();
    }
    if (t == 0) { scal[0] = smn[0]; scal[1] = smx[0]; }
}

// ---------------------------------------------------------------------------
// mat2gray + unfold (40x40 patches, stride 20) -> out[8, 1600, 2500]
// ---------------------------------------------------------------------------
__global__ void unfold_norm(const float* __restrict__ s,
                            const float* __restrict__ scal,
                            float* __restrict__ out, long total)
{
    long i = (long)blockIdx.x * blockDim.x + threadIdx.x;
    if (i >= total) return;
    const long per_b = 1600L * 2500L;
    const int  b  = (int)(i / per_b);
    const long r  = i % per_b;
    const int  ch = (int)(r / 2500);
    const int  p  = (int)(r % 2500);
    const int kh = ch / 40, kw = ch % 40;
    const int py = p / 50,  px = p % 50;
    const int y = py * 20 + kh, x = px * 20 + kw;
    const float mn = scal[0], mx = scal[1];
    const float v = s[(size_t)b * 1026 * 1026 + (size_t)y * 1026 + x];
    out[i] = (v - mn) / (mx - mn + 1.1920929e-7f);
}

// ---------------------------------------------------------------------------
// Host launcher
// ---------------------------------------------------------------------------
static void make_taps(Taps* t, double sigma) {
    double e[KTAPS];
    double s = 0.0;
    for (int i = 0; i < KTAPS; ++i) {
        const double x = (double)(i - KHALF);
        e[i] = exp(-(x * x) / sigma);
        s += e[i];
    }
    for (int i = 0; i < KTAPS; ++i) t->g[i] = (float)(e[i] / s);
}

extern "C" void kernel_launch(void* const* d_in, const int* in_sizes, int n_in,
                              void* d_out, int out_size, void* d_ws, size_t ws_size,
                              hipStream_t stream) {
    (void)in_sizes; (void)n_in; (void)out_size; (void)ws_size;
    const float* inp = (const float*)d_in[0];
    float* out = (float*)d_out;

    // Workspace: 6 ping-pong buffers sized for the largest stage.
    const size_t S = 8ull * 1026ull * 1026ull;   // floats per buffer
    float* buf0 = (float*)d_ws;                  // tmpV   -> t11
    float* buf1 = buf0 + S;                      // img    -> t12
    float* buf2 = buf1 + S;                      // gxx    -> s (eigen score)
    float* buf3 = buf2 + S;                      // gxy
    float* buf4 = buf3 + S;                      // gyy
    float* buf5 = buf4 + S;                      // t22
    float* pmin = buf5 + S;                      // [512]
    float* pmax = pmin + 512;                    // [512]
    float* scal = pmax + 512;                    // [2] = {min, max}

    Taps t1, t2;                                 // recomputed every call
    make_taps(&t1, 2.0);                         // sigma1
    make_taps(&t2, 9.0);                         // sigma2

    // 1) blur1: vertical (1024x1024 -> 1025x1024), horizontal (-> 1025x1025)
    vconv_wmma<<<dim3(64, 9, 8), 256, 0, stream>>>(inp, buf0, 1024, 1024, 1025, t1);
    hconv_wmma<<<dim3(65, 9, 8), 256, 0, stream>>>(buf0, buf1, 1025, 1024, 1025, t1);

    // 2) gradient + products on img [1025x1025]
    {
        const long total = 8L * 1025 * 1025;
        const int blocks = (int)((total + 255) / 256);
        grad_products<<<blocks, 256, 0, stream>>>(buf1, buf2, buf3, buf4, 1025, 1025, total);
    }

    // 3) blur2 vertical on the three products (1025x1025 -> 1026x1025)
    vconv_wmma<<<dim3(65, 9, 8), 256, 0, stream>>>(buf2, buf0, 1025, 1025, 1026, t2); // t11
    vconv_wmma<<<dim3(65, 9, 8), 256, 0, stream>>>(buf3, buf1, 1025, 1025, 1026, t2); // t12
    vconv_wmma<<<dim3(65, 9, 8), 256, 0, stream>>>(buf4, buf5, 1025, 1025, 1026, t2); // t22

    // 4) blur2 horizontal + eigen/corner score fused (-> s [1026x1026])
    hconv2_eigen_wmma<<<dim3(65, 9, 8), 256, 0, stream>>>(buf0, buf1, buf5, buf2,
                                                          1026, 1025, 1026, t2);

    // 5) global min/max of s
    {
        const long n = 8L * 1026 * 1026;
        reduce_minmax_1<<<512, 256, 0, stream>>>(buf2, n, pmin, pmax);
        reduce_minmax_2<<<1, 256, 0, stream>>>(pmin, pmax, scal);
    }

    // 6) normalize + unfold -> [8, 1600, 2500]
    {
        const long total = 8L * 1600 * 2500;
        const int blocks = (int)((total + 255) / 256);
        unfold_norm<<<blocks, 256, 0, stream>>>(buf2, scal, out, total);
    }
}